// DDeePC_7301444403751
// MI455X (gfx1250) — compile-verified
//
#include <hip/hip_runtime.h>

typedef __attribute__((ext_vector_type(2))) float v2f;
typedef __attribute__((ext_vector_type(8))) float v8f;

#define LGr   1453      // true LG
#define PLG   1536      // LG padded to 128
#define KN    1664      // 1585 padded to 128
#define NBATCH 8192
#define NS_ITERS 50

// LDS tile: K-pair interleaved, [kpair][2*m + parity], row pitch 288 dwords
// (288 % 64 == 32 -> the two kpair rows read by lane-halves hit disjoint
// bank halves; frag read = one aligned ds_load_b64, no repack movs).
#define LPITCH 288

// ---------------------------------------------------------------------------
// TN WMMA GEMM: C[M,N] = alpha * (A^T B) + beta * Cadd
//   A: [Kd, M] row-major, B: [Kd, N] row-major.  M,N multiples of 128,
//   Kd multiple of 16.
//   256 threads (8 wave32), 128x128 C tile, K staged 16 deep, double-buffered
//   LDS pipeline (global b128 loads for chunk i+1 issued before the WMMA
//   burst of chunk i).  Each wave computes 32x64 via 8 f32 WMMA accumulators;
//   per k-step 6 ds_load_b64 fragments feed 8 V_WMMA_F32_16X16X4_F32.
// ---------------------------------------------------------------------------
__global__ __launch_bounds__(256) void k_gemm_tn(
    const float* __restrict__ A, int lda,
    const float* __restrict__ B, int ldb,
    float* __restrict__ C, int ldc,
    const float* __restrict__ Cadd, int ldca,
    int Kd, float alpha, float beta)
{
  __shared__ __align__(16) float As[2][8][LPITCH];
  __shared__ __align__(16) float Bs[2][8][LPITCH];
  const int t    = threadIdx.x;
  const int lane = t & 31;
  const int wave = t >> 5;
  const int wr   = wave >> 1;          // 0..3 -> 32-row strip
  const int wc   = wave & 1;           // 0..1 -> 64-col strip
  const int m0   = blockIdx.y * 128;
  const int n0   = blockIdx.x * 128;
  const int hf   = lane >> 4;          // lane half -> +1 kpair (ISA K pairing)
  const int lm   = lane & 15;

  // staging: each thread owns (kpair, 4-column quad)
  const int kp  = t >> 5;              // 0..7
  const int sc4 = t & 31;              // column quad 0..31

  v8f acc[2][4];
  #pragma unroll
  for (int mt = 0; mt < 2; ++mt)
    #pragma unroll
    for (int nt = 0; nt < 4; ++nt)
      acc[mt][nt] = (v8f){};

  // ---- prologue: stage chunk 0 into buffer 0 ----
  {
    const float4 a0 = *(const float4*)(A + (size_t)(2 * kp + 0) * lda + m0 + 4 * sc4);
    const float4 a1 = *(const float4*)(A + (size_t)(2 * kp + 1) * lda + m0 + 4 * sc4);
    const float4 b0 = *(const float4*)(B + (size_t)(2 * kp + 0) * ldb + n0 + 4 * sc4);
    const float4 b1 = *(const float4*)(B + (size_t)(2 * kp + 1) * ldb + n0 + 4 * sc4);
    *(float2*)&As[0][kp][2 * (4 * sc4 + 0)] = make_float2(a0.x, a1.x);
    *(float2*)&As[0][kp][2 * (4 * sc4 + 1)] = make_float2(a0.y, a1.y);
    *(float2*)&As[0][kp][2 * (4 * sc4 + 2)] = make_float2(a0.z, a1.z);
    *(float2*)&As[0][kp][2 * (4 * sc4 + 3)] = make_float2(a0.w, a1.w);
    *(float2*)&Bs[0][kp][2 * (4 * sc4 + 0)] = make_float2(b0.x, b1.x);
    *(float2*)&Bs[0][kp][2 * (4 * sc4 + 1)] = make_float2(b0.y, b1.y);
    *(float2*)&Bs[0][kp][2 * (4 * sc4 + 2)] = make_float2(b0.z, b1.z);
    *(float2*)&Bs[0][kp][2 * (4 * sc4 + 3)] = make_float2(b0.w, b1.w);
  }
  __syncthreads();

  int buf = 0;
  for (int k0 = 0; k0 < Kd; k0 += 16) {
    // ---- issue next chunk's global loads before the WMMA burst ----
    float4 na0, na1, nb0, nb1;
    const bool has_next = (k0 + 16) < Kd;     // wave-uniform: EXEC stays all-1s
    if (has_next) {
      const int kn = k0 + 16;
      na0 = *(const float4*)(A + (size_t)(kn + 2 * kp + 0) * lda + m0 + 4 * sc4);
      na1 = *(const float4*)(A + (size_t)(kn + 2 * kp + 1) * lda + m0 + 4 * sc4);
      nb0 = *(const float4*)(B + (size_t)(kn + 2 * kp + 0) * ldb + n0 + 4 * sc4);
      nb1 = *(const float4*)(B + (size_t)(kn + 2 * kp + 1) * ldb + n0 + 4 * sc4);
    }

    // ---- compute: 4 k-steps x 8 WMMA from buffer `buf` ----
    #pragma unroll
    for (int kk = 0; kk < 16; kk += 4) {
      const int kprow = (kk >> 1) + hf;       // kpair row for this lane half
      v2f a[2], b[4];
      #pragma unroll
      for (int mt = 0; mt < 2; ++mt) {
        const float2 f = *(const float2*)&As[buf][kprow][2 * (32 * wr + 16 * mt + lm)];
        a[mt].x = f.x; a[mt].y = f.y;
      }
      #pragma unroll
      for (int nt = 0; nt < 4; ++nt) {
        const float2 f = *(const float2*)&Bs[buf][kprow][2 * (64 * wc + 16 * nt + lm)];
        b[nt].x = f.x; b[nt].y = f.y;
      }
      #pragma unroll
      for (int mt = 0; mt < 2; ++mt)
        #pragma unroll
        for (int nt = 0; nt < 4; ++nt)
          acc[mt][nt] = __builtin_amdgcn_wmma_f32_16x16x4_f32(
              false, a[mt], false, b[nt], (short)0, acc[mt][nt], false, false);
    }

    // ---- store next chunk into the alternate buffer ----
    if (has_next) {
      const int nb = buf ^ 1;
      *(float2*)&As[nb][kp][2 * (4 * sc4 + 0)] = make_float2(na0.x, na1.x);
      *(float2*)&As[nb][kp][2 * (4 * sc4 + 1)] = make_float2(na0.y, na1.y);
      *(float2*)&As[nb][kp][2 * (4 * sc4 + 2)] = make_float2(na0.z, na1.z);
      *(float2*)&As[nb][kp][2 * (4 * sc4 + 3)] = make_float2(na0.w, na1.w);
      *(float2*)&Bs[nb][kp][2 * (4 * sc4 + 0)] = make_float2(nb0.x, nb1.x);
      *(float2*)&Bs[nb][kp][2 * (4 * sc4 + 1)] = make_float2(nb0.y, nb1.y);
      *(float2*)&Bs[nb][kp][2 * (4 * sc4 + 2)] = make_float2(nb0.z, nb1.z);
      *(float2*)&Bs[nb][kp][2 * (4 * sc4 + 3)] = make_float2(nb0.w, nb1.w);
    }
    buf ^= 1;
    __syncthreads();
  }

  #pragma unroll
  for (int mt = 0; mt < 2; ++mt)
    #pragma unroll
    for (int nt = 0; nt < 4; ++nt)
      #pragma unroll
      for (int v = 0; v < 8; ++v) {
        int cm = m0 + 32 * wr + 16 * mt + v + 8 * hf;  // C layout: VGPR v -> M=v (+8 hi lanes)
        int cn = n0 + 64 * wc + 16 * nt + lm;
        float r = alpha * acc[mt][nt][v];
        if (Cadd) r += beta * Cadd[(size_t)cm * ldca + cn];
        C[(size_t)cm * ldc + cn] = r;
      }
}

// ---------------------------------------------------------------------------
// Builders / glue
// ---------------------------------------------------------------------------
// Astack rows 0..127 = Wq*Yf, 128..255 = Wr*Uf ; Bstack = [Yf ; Uf].  [256 x PLG]
__global__ void k_build_stacks(const float* __restrict__ ud, const float* __restrict__ yd,
                               const float* __restrict__ qv, const float* __restrict__ rv,
                               float* __restrict__ Astack, float* __restrict__ Bstack)
{
  int idx = blockIdx.x * blockDim.x + threadIdx.x;
  if (idx >= 256 * PLG) return;
  int row = idx / PLG, j = idx % PLG;
  float av = 0.f, bv = 0.f;
  if (j < LGr) {
    if (row < 128) { float y = yd[64 + row + 4 * j]; av = qv[row & 3] * y; bv = y; }
    else { int i = row - 128; float u = ud[64 + i + 4 * j]; av = rv[i & 3] * u; bv = u; }
  }
  Astack[idx] = av; Bstack[idx] = bv;
}

// UfT/YfT: [KN x 128], UfT[k,i] = Uf[i,k] (zero beyond LGr)
__global__ void k_build_fT(const float* __restrict__ ud, const float* __restrict__ yd,
                           float* __restrict__ UfT, float* __restrict__ YfT)
{
  int idx = blockIdx.x * blockDim.x + threadIdx.x;
  if (idx >= KN * 128) return;
  int k = idx / 128, i = idx % 128;
  float uv = 0.f, yv = 0.f;
  if (k < LGr) { uv = ud[64 + i + 4 * k]; yv = yd[64 + i + 4 * k]; }
  UfT[idx] = uv; YfT[idx] = yv;
}

__global__ void k_transpose_ref(const float* __restrict__ ref, float* __restrict__ refT)
{
  int idx = blockIdx.x * blockDim.x + threadIdx.x;
  if (idx >= 128 * NBATCH) return;
  int k = idx / NBATCH, b = idx % NBATCH;
  refT[idx] = ref[b * 128 + k];
}

__device__ __forceinline__ float acon_val(int c, int j,
                                          const float* __restrict__ ud,
                                          const float* __restrict__ yd)
{
  if (c < 64)  return ud[c + 4 * j];                 // Up
  if (c < 128) return yd[(c - 64) + 4 * j];          // Yp
  return yd[188 + (c - 128) + 4 * j];                // Yf[-P:]
}

// K = [[2H, A^T],[A, 0]] zero-padded to KN x KN (symmetric)
__global__ void k_assembleK(const float* __restrict__ H,
                            const float* __restrict__ ud, const float* __restrict__ yd,
                            float* __restrict__ Kmat)
{
  int idx = blockIdx.x * blockDim.x + threadIdx.x;
  if (idx >= KN * KN) return;
  int i = idx / KN, j = idx % KN;
  float v = 0.f;
  if (i < LGr && j < LGr)                        v = 2.f * H[(size_t)i * PLG + j];
  else if (i < LGr && j >= LGr && j < LGr + 132) v = acon_val(j - LGr, i, ud, yd);
  else if (j < LGr && i >= LGr && i < LGr + 132) v = acon_val(i - LGr, j, ud, yd);
  Kmat[idx] = v;
}

__global__ void k_init_scalar(unsigned int* s) { if (threadIdx.x == 0) s[0] = 0u; }

// max abs row sum (||K||_inf = ||K||_1 since symmetric); atomicMax on float bits (>=0)
__global__ void k_rownorm(const float* __restrict__ Kmat, unsigned int* __restrict__ s)
{
  __shared__ float red[256];
  int i = blockIdx.x;
  float acc = 0.f;
  for (int j = threadIdx.x; j < KN; j += 256) acc += fabsf(Kmat[(size_t)i * KN + j]);
  red[threadIdx.x] = acc;
  __syncthreads();
  for (int off = 128; off > 0; off >>= 1) {
    if (threadIdx.x < off) red[threadIdx.x] += red[threadIdx.x + off];
    __syncthreads();
  }
  if (threadIdx.x == 0) atomicMax(s, __float_as_uint(red[0]));
}

// X0 = K / (||K||_1 ||K||_inf)
__global__ void k_initX(const float* __restrict__ Kmat, const unsigned int* __restrict__ s,
                        float* __restrict__ X)
{
  int idx = blockIdx.x * blockDim.x + threadIdx.x;
  if (idx >= KN * KN) return;
  float sn = __uint_as_float(s[0]);
  float inv = (sn > 0.f) ? 1.f / (sn * sn) : 0.f;
  X[idx] = Kmat[idx] * inv;
}

// rhs rows [LGr, KN): u_ini^T, y_ini^T, ref[:,-4:]^T, zero pad
__global__ void k_assemble_rhs(const float* __restrict__ u_ini, const float* __restrict__ y_ini,
                               const float* __restrict__ ref, float* __restrict__ rhs)
{
  int idx = blockIdx.x * blockDim.x + threadIdx.x;
  const int rows = KN - LGr;                 // 211
  if (idx >= rows * NBATCH) return;
  int rr = idx / NBATCH, b = idx % NBATCH;
  float v = 0.f;
  if (rr < 64)        v = u_ini[b * 64 + rr];
  else if (rr < 128)  v = y_ini[b * 64 + (rr - 64)];
  else if (rr < 132)  v = ref[b * 128 + 124 + (rr - 128)];
  rhs[(size_t)(LGr + rr) * NBATCH + b] = v;
}

// g = sol[:LGr].T -> out (NBATCH x LGr), 32x32 LDS-tiled transpose
__global__ __launch_bounds__(256) void k_write_g(const float* __restrict__ sol,
                                                 float* __restrict__ g)
{
  __shared__ float tile[32][33];
  const int jt = blockIdx.y * 32;   // row (j) tile in sol
  const int bt = blockIdx.x * 32;   // batch tile
  const int tx = threadIdx.x & 31;
  const int ty = threadIdx.x >> 5;  // 0..7
  #pragma unroll
  for (int r = 0; r < 32; r += 8) {
    int j = jt + ty + r;
    tile[ty + r][tx] = (j < LGr) ? sol[(size_t)j * NBATCH + bt + tx] : 0.f;
  }
  __syncthreads();
  #pragma unroll
  for (int r = 0; r < 32; r += 8) {
    int b = bt + ty + r;
    int j = jt + tx;
    if (j < LGr) g[(size_t)b * LGr + j] = tile[tx][ty + r];
  }
}

// ---------------------------------------------------------------------------
static inline void gemm(hipStream_t s, const float* A, int lda, const float* B, int ldb,
                        float* C, int ldc, const float* Cadd, int ldca,
                        int M, int N, int Kd, float alpha, float beta)
{
  dim3 grid(N / 128, M / 128);
  k_gemm_tn<<<grid, 256, 0, s>>>(A, lda, B, ldb, C, ldc, Cadd, ldca, Kd, alpha, beta);
}

extern "C" void kernel_launch(void* const* d_in, const int* in_sizes, int n_in,
                              void* d_out, int out_size, void* d_ws, size_t ws_size,
                              hipStream_t stream)
{
  const float* ref   = (const float*)d_in[0];   // (8192,128)
  // d_in[1] = uref (unused by _forward)
  const float* u_ini = (const float*)d_in[2];   // (8192,64)
  const float* y_ini = (const float*)d_in[3];   // (8192,64)
  const float* ud    = (const float*)d_in[4];   // (1500,4) flat
  const float* yd    = (const float*)d_in[5];   // (1500,4) flat
  const float* qv    = (const float*)d_in[6];   // (4,)
  const float* rv    = (const float*)d_in[7];   // (4,)

  float* w = (float*)d_ws;
  size_t o = 0;
  float* Astack = w + o; o += (size_t)256 * PLG;
  float* Bstack = w + o; o += (size_t)256 * PLG;
  float* UfT    = w + o; o += (size_t)KN * 128;
  float* YfT    = w + o; o += (size_t)KN * 128;
  float* Hm     = w + o; o += (size_t)PLG * PLG;
  float* Kmat   = w + o; o += (size_t)KN * KN;
  float* Xa     = w + o; o += (size_t)KN * KN;
  float* Xb     = w + o; o += (size_t)KN * KN;
  float* Tb     = w + o; o += (size_t)KN * KN;
  float* refT   = w + o; o += (size_t)128 * NBATCH;
  float* rhs    = w + o; o += (size_t)KN * NBATCH;
  float* sol    = w + o; o += (size_t)KN * NBATCH;
  unsigned int* scal = (unsigned int*)(w + o);

  float* out_g = (float*)d_out;                                   // 8192 x 1453
  float* out_u = out_g + (size_t)NBATCH * LGr;                    // 8192 x 128
  float* out_y = out_u + (size_t)NBATCH * 128;                    // 8192 x 128

  auto blks = [](long long n) { return (unsigned)((n + 255) / 256); };

  // 1) Hankel operands
  k_build_stacks<<<blks(256LL * PLG), 256, 0, stream>>>(ud, yd, qv, rv, Astack, Bstack);
  k_build_fT<<<blks((long long)KN * 128), 256, 0, stream>>>(ud, yd, UfT, YfT);
  k_transpose_ref<<<blks(128LL * NBATCH), 256, 0, stream>>>(ref, refT);

  // 2) H = Yf^T Wq Yf + Uf^T Wr Uf   (single K=256 stacked GEMM)
  gemm(stream, Astack, PLG, Bstack, PLG, Hm, PLG, nullptr, 0, PLG, PLG, 256, 1.f, 0.f);

  // 3) KKT matrix, norm, X0
  k_assembleK<<<blks((long long)KN * KN), 256, 0, stream>>>(Hm, ud, yd, Kmat);
  k_init_scalar<<<1, 32, 0, stream>>>(scal);
  k_rownorm<<<KN, 256, 0, stream>>>(Kmat, scal);
  k_initX<<<blks((long long)KN * KN), 256, 0, stream>>>(Kmat, scal, Xa);

  // 4) Newton–Schulz pinv: X <- 2X - X K X  (all TN GEMMs; K, X symmetric)
  float* Xc = Xa; float* Xn = Xb;
  for (int it = 0; it < NS_ITERS; ++it) {
    gemm(stream, Kmat, KN, Xc, KN, Tb, KN, nullptr, 0, KN, KN, KN, 1.f, 0.f);   // T = K X
    gemm(stream, Xc, KN, Tb, KN, Xn, KN, Xc, KN, KN, KN, KN, -1.f, 2.f);        // X' = 2X - X T
    float* tmp = Xc; Xc = Xn; Xn = tmp;
  }

  // 5) rhs: rows [0,LGr) = 2 * Yfw^T ref^T ; rows [LGr,KN) from inis / ref tail
  gemm(stream, Astack, PLG, refT, NBATCH, rhs, NBATCH, nullptr, 0, PLG, NBATCH, 128, 2.f, 0.f);
  k_assemble_rhs<<<blks((long long)(KN - LGr) * NBATCH), 256, 0, stream>>>(u_ini, y_ini, ref, rhs);

  // 6) sol = pinv(K) rhs
  gemm(stream, Xc, KN, rhs, NBATCH, sol, NBATCH, nullptr, 0, KN, NBATCH, KN, 1.f, 0.f);

  // 7) outputs: g, u = g Uf^T, y = g Yf^T
  {
    dim3 gt(NBATCH / 32, (LGr + 31) / 32);
    k_write_g<<<gt, 256, 0, stream>>>(sol, out_g);
  }
  gemm(stream, sol, NBATCH, UfT, 128, out_u, 128, nullptr, 0, NBATCH, 128, KN, 1.f, 0.f);
  gemm(stream, sol, NBATCH, YfT, 128, out_y, 128, nullptr, 0, NBATCH, 128, KN, 1.f, 0.f);
}